// AttnBlock_29222957482472
// MI455X (gfx1250) — compile-verified
//
#include <hip/hip_runtime.h>

// MI455X / gfx1250: wave32, WMMA (V_WMMA_F32_16X16X4_F32 for exact fp32 MMA).
//
// Shapes: B=4, H=W=D=C=16, N=H*W*D=4096.
// Scrambled-attention identity (derived from the reference's transposes):
//   Q[b][(w,d,c)][h] = q[b,h,w,d,c]   (same for K, V)
//   scores = Q @ K^T  (contraction over h, depth 16), softmax(scale=1/4)
//   attn   = P @ V    (N x 16)
//   out[b,c',h',w',c''] = x[...] + sum_d' attn[b,(h',w',d'),c'] * wp[d'][c''] + bp[c'']
//
// Kernel C processes 32 score columns per online-softmax update: two S tiles
// share one running-max/sum pass, halving the cross-lane shuffle reductions
// per column (the non-WMMA bottleneck of the inner loop).

typedef __attribute__((ext_vector_type(2))) float v2f;
typedef __attribute__((ext_vector_type(8))) float v8f;

__device__ __forceinline__ v8f wmma_f32(v2f a, v2f b, v8f c) {
  // (neg_a, A, neg_b, B, c_mod, C, reuse_a, reuse_b)
  return __builtin_amdgcn_wmma_f32_16x16x4_f32(false, a, false, b, (short)0, c,
                                               false, false);
}

// ---------------------------------------------------------------------------
// Kernel A: per-(b,c) mean / rstd over the 4096 spatial positions.
// ---------------------------------------------------------------------------
__global__ void stats_kernel(const float* __restrict__ x,
                             float* __restrict__ stats) {
  __shared__ float s1[256];
  __shared__ float s2[256];
  const int bc = blockIdx.x;           // 0..63
  const int b = bc >> 4, c = bc & 15;
  const float* xb = x + b * 65536 + c; // stride 16 over spatial index
  float sum = 0.f, sq = 0.f;
  for (int s = threadIdx.x; s < 4096; s += 256) {
    float v = xb[s * 16];
    sum += v;
    sq += v * v;
  }
  s1[threadIdx.x] = sum;
  s2[threadIdx.x] = sq;
  __syncthreads();
  for (int off = 128; off > 0; off >>= 1) {
    if (threadIdx.x < off) {
      s1[threadIdx.x] += s1[threadIdx.x + off];
      s2[threadIdx.x] += s2[threadIdx.x + off];
    }
    __syncthreads();
  }
  if (threadIdx.x == 0) {
    float mean = s1[0] * (1.f / 4096.f);
    float var  = s2[0] * (1.f / 4096.f) - mean * mean;
    stats[bc]      = mean;
    stats[64 + bc] = rsqrtf(var + 1e-5f);
  }
}

// ---------------------------------------------------------------------------
// Kernel B: normalized QKV projections into scrambled (B, N, 16) layouts.
// One wave handles tile (b, h, w) : 16 voxels along d.
// A operand = normalized x (16 voxels x 16 in-channels), B operand = weight.
// ---------------------------------------------------------------------------
__global__ void qkv_kernel(const float* __restrict__ x,
                           const float* __restrict__ wq, const float* __restrict__ bq,
                           const float* __restrict__ wk, const float* __restrict__ bk,
                           const float* __restrict__ wv, const float* __restrict__ bv,
                           const float* __restrict__ stats,
                           float* __restrict__ Qm, float* __restrict__ Km,
                           float* __restrict__ Vm) {
  const int lane = threadIdx.x & 31;
  const int wvid = threadIdx.x >> 5;
  const int tile = blockIdx.x * 8 + wvid;   // 0..1023
  const int b  = tile >> 8;
  const int hh = (tile >> 4) & 15;
  const int w  = tile & 15;
  const int hi = lane >> 4;                  // lane half (0/1)
  const int ln = lane & 15;

  const float* mean = stats;
  const float* rstd = stats + 64;
  const float* xb = x + ((b * 16 + hh) * 16 + w) * 256;  // + d*16 + c'

  // A layout (16x4 f32 per chunk): lane row M=d=ln, VGPR pair = K = 4k+2*hi, +1
  v2f ha[4];
#pragma unroll
  for (int k = 0; k < 4; ++k) {
    const int c0 = 4 * k + 2 * hi;
    const float m0 = mean[b * 16 + c0],     r0 = rstd[b * 16 + c0];
    const float m1 = mean[b * 16 + c0 + 1], r1 = rstd[b * 16 + c0 + 1];
    ha[k].x = (xb[ln * 16 + c0]     - m0) * r0;
    ha[k].y = (xb[ln * 16 + c0 + 1] - m1) * r1;
  }

  v8f cq, ck, cv;
#pragma unroll
  for (int r = 0; r < 8; ++r) { cq[r] = 0.f; ck[r] = 0.f; cv[r] = 0.f; }

#pragma unroll
  for (int k = 0; k < 4; ++k) {
    const int r0 = 4 * k + 2 * hi;   // contraction rows in this chunk
    v2f bb;
    bb.x = wq[r0 * 16 + ln];       bb.y = wq[(r0 + 1) * 16 + ln];
    cq = wmma_f32(ha[k], bb, cq);
    bb.x = wk[r0 * 16 + ln];       bb.y = wk[(r0 + 1) * 16 + ln];
    ck = wmma_f32(ha[k], bb, ck);
    bb.x = wv[r0 * 16 + ln];       bb.y = wv[(r0 + 1) * 16 + ln];
    cv = wmma_f32(ha[k], bb, cv);
  }

  const float bqv = bq[ln], bkv = bk[ln], bvv = bv[ln];
#pragma unroll
  for (int r = 0; r < 8; ++r) {
    const int d   = r + 8 * hi;              // C-layout row = voxel d
    const int row = w * 256 + d * 16 + ln;   // scrambled n = (w,d,c)
    const int idx = (b << 16) + row * 16 + hh;
    Qm[idx] = (cq[r] + bqv) * 0.25f;         // fold logits scale 1/sqrt(16)
    Km[idx] = ck[r] + bkv;
    Vm[idx] = cv[r] + bvv;
  }
}

// ---------------------------------------------------------------------------
// Kernel C: flash attention (online softmax, 32 columns/step) + output
// projection + residual. One wave per 16-row tile (fixed b, h', w').
// ---------------------------------------------------------------------------
__global__ void attn_kernel(const float* __restrict__ x,
                            const float* __restrict__ Qm,
                            const float* __restrict__ Km,
                            const float* __restrict__ Vm,
                            const float* __restrict__ wp,
                            const float* __restrict__ bp,
                            float* __restrict__ out) {
  __shared__ float lds[8 * 512];             // 16x32 f32 staging per wave
  const int lane = threadIdx.x & 31;
  const int wvid = threadIdx.x >> 5;
  float* P = lds + wvid * 512;

  const int tile = blockIdx.x * 8 + wvid;    // 0..1023
  const int b  = tile >> 8;
  const int t  = tile & 255;                 // row tile within batch
  const int n0 = t * 16;
  const int hp = t >> 4, wp_ = t & 15;       // (h', w') for the projection
  const int hi = lane >> 4;
  const int ln = lane & 15;

  const float* Qb = Qm + (b << 16);
  const float* Kb = Km + (b << 16);
  const float* Vb = Vm + (b << 16);

  // Q tile in A layout (reused for all column tiles).
  v2f qa[4];
#pragma unroll
  for (int k = 0; k < 4; ++k) {
    const float* qrow = Qb + (n0 + ln) * 16 + 4 * k + 2 * hi;
    qa[k].x = qrow[0];
    qa[k].y = qrow[1];
  }

  v8f o;
  float mrun[8], lrun[8];
#pragma unroll
  for (int r = 0; r < 8; ++r) { o[r] = 0.f; mrun[r] = -1e30f; lrun[r] = 0.f; }

  for (int m0 = 0; m0 < 4096; m0 += 32) {
    // Two 16x16 S tiles: columns [m0, m0+16) and [m0+16, m0+32).
    v8f s0, s1;
#pragma unroll
    for (int r = 0; r < 8; ++r) { s0[r] = 0.f; s1[r] = 0.f; }
#pragma unroll
    for (int k = 0; k < 4; ++k) {
      const float* krow0 = Kb + (m0 + ln) * 16 + 4 * k + 2 * hi;
      const float* krow1 = krow0 + 256;      // (m0+16+ln)*16
      v2f kb;
      kb.x = krow0[0];
      kb.y = krow0[1];
      s0 = wmma_f32(qa[k], kb, s0);
      kb.x = krow1[0];
      kb.y = krow1[1];
      s1 = wmma_f32(qa[k], kb, s1);
    }

    // Online softmax; C-layout row = r + 8*hi, column = ln (across 16 lanes).
    // One running-max/sum update covers both tiles (32 columns).
#pragma unroll
    for (int r = 0; r < 8; ++r) {
      const float sv0 = s0[r];
      const float sv1 = s1[r];
      float rmax = fmaxf(sv0, sv1);
      rmax = fmaxf(rmax, __shfl_xor(rmax, 1));
      rmax = fmaxf(rmax, __shfl_xor(rmax, 2));
      rmax = fmaxf(rmax, __shfl_xor(rmax, 4));
      rmax = fmaxf(rmax, __shfl_xor(rmax, 8));
      const float mnew = fmaxf(mrun[r], rmax);
      const float corr = __expf(mrun[r] - mnew);
      const float p0   = __expf(sv0 - mnew);
      const float p1   = __expf(sv1 - mnew);
      float rsum = p0 + p1;
      rsum += __shfl_xor(rsum, 1);
      rsum += __shfl_xor(rsum, 2);
      rsum += __shfl_xor(rsum, 4);
      rsum += __shfl_xor(rsum, 8);
      lrun[r] = lrun[r] * corr + rsum;
      mrun[r] = mnew;
      o[r] *= corr;
      const int row = (r + 8 * hi) * 32;
      P[row + ln]      = p0;                 // stage P row-major [n][m] 16x32
      P[row + 16 + ln] = p1;
    }
    asm volatile("s_wait_dscnt 0" ::: "memory");  // same-wave LDS RAW

    // O += P @ V : reload P in A layout (K = 32 -> 8 chunks of 4).
#pragma unroll
    for (int k = 0; k < 8; ++k) {
      v2f pa;
      pa.x = P[ln * 32 + 4 * k + 2 * hi];
      pa.y = P[ln * 32 + 4 * k + 2 * hi + 1];
      const int mr = m0 + 4 * k + 2 * hi;
      v2f vb;
      vb.x = Vb[mr * 16 + ln];
      vb.y = Vb[(mr + 1) * 16 + ln];
      o = wmma_f32(pa, vb, o);
    }
  }

  // Finalize softmax and stage O as [d'][c'] for the transposed projection.
#pragma unroll
  for (int r = 0; r < 8; ++r)
    P[(r + 8 * hi) * 16 + ln] = o[r] / lrun[r];
  asm volatile("s_wait_dscnt 0" ::: "memory");

  // R[c'][c''] = sum_d' O[d'][c'] * wp[d'][c'']   (A = O^T via LDS transpose)
  v8f R;
#pragma unroll
  for (int r = 0; r < 8; ++r) R[r] = 0.f;
#pragma unroll
  for (int k = 0; k < 4; ++k) {
    const int d0 = 4 * k + 2 * hi;
    v2f oa;
    oa.x = P[d0 * 16 + ln];
    oa.y = P[(d0 + 1) * 16 + ln];
    v2f wb;
    wb.x = wp[d0 * 16 + ln];
    wb.y = wp[(d0 + 1) * 16 + ln];
    R = wmma_f32(oa, wb, R);
  }

  // y[b, c', h', w', c''] = x[same] + R + bp[c'']   (coalesced 16-float rows)
  const float bpv = bp[ln];
#pragma unroll
  for (int r = 0; r < 8; ++r) {
    const int cp  = r + 8 * hi;
    const int idx = (b << 16) + (cp << 12) + (hp << 8) + (wp_ << 4) + ln;
    out[idx] = x[idx] + R[r] + bpv;
  }
}

// ---------------------------------------------------------------------------
extern "C" void kernel_launch(void* const* d_in, const int* in_sizes, int n_in,
                              void* d_out, int out_size, void* d_ws,
                              size_t ws_size, hipStream_t stream) {
  const float* x  = (const float*)d_in[0];
  const float* wq = (const float*)d_in[1];
  const float* bq = (const float*)d_in[2];
  const float* wk = (const float*)d_in[3];
  const float* bk = (const float*)d_in[4];
  const float* wv = (const float*)d_in[5];
  const float* bv = (const float*)d_in[6];
  const float* wp = (const float*)d_in[7];
  const float* bp = (const float*)d_in[8];
  float* out = (float*)d_out;

  float* ws    = (float*)d_ws;
  float* stats = ws;                 // 128 floats (mean[64], rstd[64])
  float* Qm    = ws + 128;           // 4*4096*16
  float* Km    = Qm + 262144;
  float* Vm    = Km + 262144;

  stats_kernel<<<64, 256, 0, stream>>>(x, stats);
  qkv_kernel<<<128, 256, 0, stream>>>(x, wq, bq, wk, bk, wv, bv, stats,
                                      Qm, Km, Vm);
  attn_kernel<<<128, 256, 0, stream>>>(x, Qm, Km, Vm, wp, bp, out);
}